// BaseLayer_91130616086688
// MI455X (gfx1250) — compile-verified
//
#include <hip/hip_runtime.h>
#include <hip/hip_bf16.h>

typedef float v2f __attribute__((ext_vector_type(2)));
typedef float v8f __attribute__((ext_vector_type(8)));

// ---------------------------------------------------------------------------
// Kernel 1: embeddings + static features -> base_input[:, 0:48]
// base_input row = [x_static(16) | time_emb(16) | day_emb(16) | gru_h(64)] = 112
// ---------------------------------------------------------------------------
__global__ void embed_kernel(const float* __restrict__ xs,
                             const int* __restrict__ tidx,
                             const int* __restrict__ didx,
                             const float* __restrict__ temb,
                             const float* __restrict__ demb,
                             float* __restrict__ base, int N) {
  int n = blockIdx.x * blockDim.x + threadIdx.x;
  if (n >= N) return;
  const float4* xsr = (const float4*)(xs + (size_t)n * 16);
  float4* br = (float4*)(base + (size_t)n * 112);
  br[0] = xsr[0]; br[1] = xsr[1]; br[2] = xsr[2]; br[3] = xsr[3];
  const float4* tr = (const float4*)(temb + (size_t)tidx[n] * 16);
  br[4] = tr[0]; br[5] = tr[1]; br[6] = tr[2]; br[7] = tr[3];
  const float4* dr = (const float4*)(demb + (size_t)didx[n] * 16);
  br[8] = dr[0]; br[9] = dr[1]; br[10] = dr[2]; br[11] = dr[3];
}

// ---------------------------------------------------------------------------
// Kernel 2: GRU over T=12, hidden 64, scalar input.
// One wave handles 16 nodes. h kept as 4 C-layout WMMA fragments (tiles 0..3).
// gh = h @ w_hh^T computed as 12 16x16 tiles with V_WMMA_F32_16X16X4_F32;
// tiles (t, t+4, t+8) give (r,z,n) for the SAME (node, hidden) per lane/vgpr,
// so gate math is pure per-lane VALU. LDS round-trip converts h to A-layout.
// ---------------------------------------------------------------------------
#define GRU_WAVES 4
__global__ __launch_bounds__(32 * GRU_WAVES) void gru_kernel(
    const float* __restrict__ hist, const float* __restrict__ w_ih,
    const float* __restrict__ w_hh, const float* __restrict__ b_ih,
    const float* __restrict__ b_hh, float* __restrict__ base, int N) {
  __shared__ float lh[GRU_WAVES][16 * 64];   // h in row-major [m][k]
  __shared__ float lx[GRU_WAVES][12 * 16];   // xt tile [t][m]
  const int wave = threadIdx.x >> 5;
  const int lane = threadIdx.x & 31;
  const int lo = lane & 15, hi = lane >> 4;
  const int nodeBase = (blockIdx.x * GRU_WAVES + wave) * 16;

  // preload xt tile (zero-padded past N), zero-init h in LDS
  for (int p = lane; p < 12 * 16; p += 32) {
    int m = p & 15, tt = p >> 4;
    float v = 0.f;
    if (nodeBase + m < N) v = hist[(size_t)(nodeBase + m) * 12 + tt];
    lx[wave][tt * 16 + m] = v;
  }
  for (int p = lane; p < 16 * 64; p += 32) lh[wave][p] = 0.f;

  // per-lane constants: input weights/biases for hidden j = t*16+lo
  float wr[4], wz[4], wn[4], br_[4], bz_[4], bn_[4], bhh[12];
  for (int t = 0; t < 4; ++t) {
    int j = t * 16 + lo;
    wr[t] = w_ih[j];        wz[t] = w_ih[64 + j];  wn[t] = w_ih[128 + j];
    br_[t] = b_ih[j];       bz_[t] = b_ih[64 + j]; bn_[t] = b_ih[128 + j];
  }
  for (int t = 0; t < 12; ++t) bhh[t] = b_hh[t * 16 + lo];

  float hc[4][8];
  for (int t = 0; t < 4; ++t)
    for (int v = 0; v < 8; ++v) hc[t][v] = 0.f;
  __syncthreads();

  for (int step = 0; step < 12; ++step) {
    // A-fragments for all 16 K-slabs: lane holds M=lo, K = kk*4 + 2*hi + {0,1}
    v2f afrag[16];
    for (int kk = 0; kk < 16; ++kk)
      afrag[kk] = *(const v2f*)&lh[wave][lo * 64 + kk * 4 + 2 * hi];
    float xtv[8];
    for (int v = 0; v < 8; ++v) xtv[v] = lx[wave][step * 16 + v + 8 * hi];

    for (int t = 0; t < 4; ++t) {
      v8f cr, cz, cn;
      for (int v = 0; v < 8; ++v) { cr[v] = bhh[t]; cz[v] = bhh[t + 4]; cn[v] = bhh[t + 8]; }
      for (int kk = 0; kk < 16; ++kk) {
        // B frag: col j = tile*16+lo, rows K = kk*4 + 2*hi + {0,1};
        // B[k,n] = w_hh[n_out, k] -> consecutive k contiguous in w_hh row
        v2f brf = *(const v2f*)&w_hh[(size_t)((t    ) * 16 + lo) * 64 + kk * 4 + 2 * hi];
        v2f bzf = *(const v2f*)&w_hh[(size_t)((t + 4) * 16 + lo) * 64 + kk * 4 + 2 * hi];
        v2f bnf = *(const v2f*)&w_hh[(size_t)((t + 8) * 16 + lo) * 64 + kk * 4 + 2 * hi];
        cr = __builtin_amdgcn_wmma_f32_16x16x4_f32(false, afrag[kk], false, brf, (short)0, cr, false, false);
        cz = __builtin_amdgcn_wmma_f32_16x16x4_f32(false, afrag[kk], false, bzf, (short)0, cz, false, false);
        cn = __builtin_amdgcn_wmma_f32_16x16x4_f32(false, afrag[kk], false, bnf, (short)0, cn, false, false);
      }
      for (int v = 0; v < 8; ++v) {
        float gi_r = fmaf(xtv[v], wr[t], br_[t]);
        float gi_z = fmaf(xtv[v], wz[t], bz_[t]);
        float gi_n = fmaf(xtv[v], wn[t], bn_[t]);
        float r = 1.f / (1.f + __expf(-(gi_r + cr[v])));
        float z = 1.f / (1.f + __expf(-(gi_z + cz[v])));
        float nc = tanhf(gi_n + r * cn[v]);
        hc[t][v] = (1.f - z) * nc + z * hc[t][v];
      }
    }
    // write h back to LDS (row-major) for next step's A-fragments
    for (int t = 0; t < 4; ++t)
      for (int v = 0; v < 8; ++v)
        lh[wave][(v + 8 * hi) * 64 + t * 16 + lo] = hc[t][v];
  }
  // temporal feature -> base_input[:, 48:112]
  for (int t = 0; t < 4; ++t)
    for (int v = 0; v < 8; ++v) {
      int m = nodeBase + v + 8 * hi;
      if (m < N) base[(size_t)m * 112 + 48 + t * 16 + lo] = hc[t][v];
    }
}

// ---------------------------------------------------------------------------
// Kernel 3: generic WMMA f32 matmul C[M,Ncols] = A[M,K] @ B[K,Ncols]
// block = 8 waves; one 16-row M-tile per block, A staged in LDS; waves stripe
// over N-tiles. K % 4 == 0, Ncols % 16 == 0, K <= 112.
// ---------------------------------------------------------------------------
__global__ __launch_bounds__(256) void wmma_matmul_kernel(
    const float* __restrict__ A, const float* __restrict__ B,
    float* __restrict__ C, int M, int K, int Ncols) {
  __shared__ float lA[16 * 112];
  const int tid = threadIdx.x;
  const int mbase = blockIdx.x * 16;
  for (int idx = tid; idx < 16 * K; idx += 256) {
    int m = idx / K, k = idx - m * K;
    lA[idx] = (mbase + m < M) ? A[(size_t)(mbase + m) * K + k] : 0.f;
  }
  __syncthreads();
  const int wave = tid >> 5, lane = tid & 31;
  const int lo = lane & 15, hi = lane >> 4;
  const int ntiles = Ncols >> 4;
  for (int nt = wave; nt < ntiles; nt += 8) {
    v8f acc;
    for (int v = 0; v < 8; ++v) acc[v] = 0.f;
    for (int kk = 0; kk < (K >> 2); ++kk) {
      v2f a = *(const v2f*)&lA[lo * K + kk * 4 + 2 * hi];
      int krow = kk * 4 + 2 * hi;
      v2f b;
      b.x = B[(size_t)krow * Ncols + nt * 16 + lo];
      b.y = B[(size_t)(krow + 1) * Ncols + nt * 16 + lo];
      acc = __builtin_amdgcn_wmma_f32_16x16x4_f32(false, a, false, b, (short)0, acc, false, false);
    }
    for (int v = 0; v < 8; ++v) {
      int m = mbase + v + 8 * hi;
      if (m < M) C[(size_t)m * Ncols + nt * 16 + lo] = acc[v];
    }
  }
}

// ---------------------------------------------------------------------------
// Attention coefficients: es/ed[n,h] = <feat[n,h,:], a_src/dst[h,:]>
// ---------------------------------------------------------------------------
__global__ void attn_kernel(const float* __restrict__ feat,
                            const float* __restrict__ asrc,
                            const float* __restrict__ adst,
                            float* __restrict__ es, float* __restrict__ ed,
                            int N, int H) {
  int t = blockIdx.x * blockDim.x + threadIdx.x;
  if (t >= N * H) return;
  int n = t / H, h = t - n * H;
  const float* f = feat + (size_t)(n * H + h) * 64;
  const float* as_ = asrc + h * 64;
  const float* ad_ = adst + h * 64;
  float ss = 0.f, dd = 0.f;
  for (int c = 0; c < 64; ++c) { float x = f[c]; ss = fmaf(x, as_[c], ss); dd = fmaf(x, ad_[c], dd); }
  es[t] = ss; ed[t] = dd;
}

// ---------------------------------------------------------------------------
// Edge softmax machinery. Segment-max via monotonic uint encoding + atomicMax.
// enc is order-preserving float->uint; init value 0u is below any real encode.
// ---------------------------------------------------------------------------
__device__ __forceinline__ unsigned enc_f(float x) {
  unsigned u = __float_as_uint(x);
  return (u & 0x80000000u) ? ~u : (u | 0x80000000u);
}
__device__ __forceinline__ float dec_f(unsigned k) {
  return __uint_as_float((k & 0x80000000u) ? (k ^ 0x80000000u) : ~k);
}
__device__ __forceinline__ void edge_sd(const int* ei, int E, int e, int& src, int& dst) {
  if (e < E) { src = ei[e]; dst = ei[E + e]; }
  else       { src = e - E; dst = e - E; }       // self loops
}

__global__ void fill_u32_kernel(unsigned* __restrict__ p, unsigned v, int n) {
  int t = blockIdx.x * blockDim.x + threadIdx.x;
  if (t < n) p[t] = v;
}

__global__ void edge_max_kernel(const float* __restrict__ es, const float* __restrict__ ed,
                                const int* __restrict__ ei, int E, int N, int H,
                                unsigned* __restrict__ mx) {
  int e = blockIdx.x * blockDim.x + threadIdx.x;
  if (e >= E + N) return;
  int src, dst; edge_sd(ei, E, e, src, dst);
  for (int h = 0; h < H; ++h) {
    float x = es[src * H + h] + ed[dst * H + h];
    x = (x > 0.f) ? x : 0.2f * x;                // leaky_relu 0.2
    atomicMax(&mx[dst * H + h], enc_f(x));
  }
}

__global__ void edge_expsum_kernel(const float* __restrict__ es, const float* __restrict__ ed,
                                   const int* __restrict__ ei, int E, int N, int H,
                                   const unsigned* __restrict__ mx,
                                   float* __restrict__ ex, float* __restrict__ s) {
  int e = blockIdx.x * blockDim.x + threadIdx.x;
  if (e >= E + N) return;
  int src, dst; edge_sd(ei, E, e, src, dst);
  for (int h = 0; h < H; ++h) {
    float x = es[src * H + h] + ed[dst * H + h];
    x = (x > 0.f) ? x : 0.2f * x;
    float v = __expf(x - dec_f(mx[dst * H + h]));
    ex[(size_t)e * H + h] = v;
    atomicAdd(&s[dst * H + h], v);
  }
}

__global__ void edge_scatter_kernel(const float* __restrict__ feat,
                                    const int* __restrict__ ei, int E, int N, int H,
                                    const float* __restrict__ ex, const float* __restrict__ s,
                                    float* __restrict__ acc) {
  long long t = (long long)blockIdx.x * blockDim.x + threadIdx.x;
  long long total = (long long)(E + N) * 64;
  if (t >= total) return;
  int e = (int)(t >> 6), c = (int)(t & 63);
  int src, dst; edge_sd(ei, E, e, src, dst);
  for (int h = 0; h < H; ++h) {
    float alpha = ex[(size_t)e * H + h] / s[dst * H + h];
    atomicAdd(&acc[(size_t)(dst * H + h) * 64 + c],
              alpha * feat[(size_t)(src * H + h) * 64 + c]);
  }
}

// out[n,c] = relu( mean_h(acc[n,h,c]) + bias[c] )
__global__ void combine_kernel(const float* __restrict__ acc, const float* __restrict__ bias,
                               float* __restrict__ out, int N, int H) {
  int t = blockIdx.x * blockDim.x + threadIdx.x;
  if (t >= N * 64) return;
  int n = t >> 6, c = t & 63;
  float v = 0.f;
  for (int h = 0; h < H; ++h) v += acc[(size_t)(n * H + h) * 64 + c];
  v = v * (1.f / (float)H) + bias[c];
  out[t] = v > 0.f ? v : 0.f;
}

// ---------------------------------------------------------------------------
// Host orchestration
// ---------------------------------------------------------------------------
static inline int cdiv(long long a, long long b) { return (int)((a + b - 1) / b); }

extern "C" void kernel_launch(void* const* d_in, const int* in_sizes, int n_in,
                              void* d_out, int out_size, void* d_ws, size_t ws_size,
                              hipStream_t stream) {
  const int N = in_sizes[0] / 16;        // 50000
  const int E = in_sizes[4] / 2;         // 800000
  const int EP = E + N;                  // with self loops

  const float* xs    = (const float*)d_in[0];
  const float* hist  = (const float*)d_in[1];
  const int*   tidx  = (const int*)d_in[2];
  const int*   didx  = (const int*)d_in[3];
  const int*   ei    = (const int*)d_in[4];
  const float* temb  = (const float*)d_in[5];
  const float* demb  = (const float*)d_in[6];
  const float* w_ih  = (const float*)d_in[7];
  const float* w_hh  = (const float*)d_in[8];
  const float* b_ih  = (const float*)d_in[9];
  const float* b_hh  = (const float*)d_in[10];
  const float* W1    = (const float*)d_in[11];
  const float* a1s   = (const float*)d_in[12];
  const float* a1d   = (const float*)d_in[13];
  const float* b1    = (const float*)d_in[14];
  const float* W2    = (const float*)d_in[15];
  const float* a2s   = (const float*)d_in[16];
  const float* a2d   = (const float*)d_in[17];
  const float* b2    = (const float*)d_in[18];
  float* outp = (float*)d_out;

  // workspace layout (floats); layer 2 reuses layer-1 regions
  float* ws = (float*)d_ws;
  size_t o = 0;
  float*    base_input = ws + o; o += (size_t)N * 112;   // reused as h2 in layer 2
  float*    h1         = ws + o; o += (size_t)N * 256;
  float*    es         = ws + o; o += (size_t)N * 4;
  float*    ed         = ws + o; o += (size_t)N * 4;
  unsigned* mx  = (unsigned*)(ws + o); o += (size_t)N * 4;
  float*    s          = ws + o; o += (size_t)N * 4;
  float*    ex         = ws + o; o += (size_t)EP * 4;
  float*    acc        = ws + o; o += (size_t)N * 256;
  float*    out1       = ws + o; o += (size_t)N * 64;
  (void)o; (void)ws_size; (void)n_in; (void)out_size;

  const int B = 256;

  // ---- features ----
  embed_kernel<<<cdiv(N, B), B, 0, stream>>>(xs, tidx, didx, temb, demb, base_input, N);
  gru_kernel<<<cdiv(N, 16 * GRU_WAVES), 32 * GRU_WAVES, 0, stream>>>(
      hist, w_ih, w_hh, b_ih, b_hh, base_input, N);

  // ---- GAT layer 1 (H=4) ----
  wmma_matmul_kernel<<<cdiv(N, 16), 256, 0, stream>>>(base_input, W1, h1, N, 112, 256);
  attn_kernel<<<cdiv((long long)N * 4, B), B, 0, stream>>>(h1, a1s, a1d, es, ed, N, 4);
  fill_u32_kernel<<<cdiv((long long)N * 4, B), B, 0, stream>>>(mx, 0u, N * 4);
  fill_u32_kernel<<<cdiv((long long)N * 4, B), B, 0, stream>>>((unsigned*)s, 0u, N * 4);
  fill_u32_kernel<<<cdiv((long long)N * 256, B), B, 0, stream>>>((unsigned*)acc, 0u, N * 256);
  edge_max_kernel<<<cdiv(EP, B), B, 0, stream>>>(es, ed, ei, E, N, 4, mx);
  edge_expsum_kernel<<<cdiv(EP, B), B, 0, stream>>>(es, ed, ei, E, N, 4, mx, ex, s);
  edge_scatter_kernel<<<cdiv((long long)EP * 64, B), B, 0, stream>>>(h1, ei, E, N, 4, ex, s, acc);
  combine_kernel<<<cdiv((long long)N * 64, B), B, 0, stream>>>(acc, b1, out1, N, 4);

  // ---- GAT layer 2 (H=1) ----
  float* h2 = base_input;  // reuse
  wmma_matmul_kernel<<<cdiv(N, 16), 256, 0, stream>>>(out1, W2, h2, N, 64, 64);
  attn_kernel<<<cdiv(N, B), B, 0, stream>>>(h2, a2s, a2d, es, ed, N, 1);
  fill_u32_kernel<<<cdiv(N, B), B, 0, stream>>>(mx, 0u, N);
  fill_u32_kernel<<<cdiv(N, B), B, 0, stream>>>((unsigned*)s, 0u, N);
  fill_u32_kernel<<<cdiv((long long)N * 64, B), B, 0, stream>>>((unsigned*)acc, 0u, N * 64);
  edge_max_kernel<<<cdiv(EP, B), B, 0, stream>>>(es, ed, ei, E, N, 1, mx);
  edge_expsum_kernel<<<cdiv(EP, B), B, 0, stream>>>(es, ed, ei, E, N, 1, mx, ex, s);
  edge_scatter_kernel<<<cdiv((long long)EP * 64, B), B, 0, stream>>>(h2, ei, E, N, 1, ex, s, acc);
  combine_kernel<<<cdiv((long long)N * 64, B), B, 0, stream>>>(acc, b2, outp, N, 1);
}